// NnHalfKACuda_36498632081981
// MI455X (gfx1250) — compile-verified
//
#include <hip/hip_runtime.h>
#include <math.h>

#ifndef __has_builtin
#define __has_builtin(x) 0
#endif

#define K_FEAT 32
#define FT_OUT 512
#define N_FFT  768
#define WAVES_PER_BLOCK 8

typedef int v4i __attribute__((ext_vector_type(4)));
typedef __attribute__((address_space(1))) v4i gv4i_t;   // global (AS1) v4i32
typedef __attribute__((address_space(3))) v4i lv4i_t;   // LDS (AS3) v4i32

static __device__ __forceinline__ void fma4(float4& a, float s, const float4 w) {
    a.x = fmaf(s, w.x, a.x);
    a.y = fmaf(s, w.y, a.y);
    a.z = fmaf(s, w.z, a.z);
    a.w = fmaf(s, w.w, a.w);
}

static __device__ __forceinline__ float4 clamp01_4(float4 v) {
    v.x = fminf(fmaxf(v.x, 0.0f), 1.0f);
    v.y = fminf(fmaxf(v.y, 0.0f), 1.0f);
    v.z = fminf(fmaxf(v.z, 0.0f), 1.0f);
    v.w = fminf(fmaxf(v.w, 0.0f), 1.0f);
    return v;
}

static __device__ __forceinline__ float dot4(const float4 a, const float4 b) {
    return fmaf(a.x, b.x, fmaf(a.y, b.y, fmaf(a.z, b.z, a.w * b.w)));
}

static __device__ __forceinline__ float wave32_reduce_add(float v) {
    v += __shfl_xor(v, 16, 32);
    v += __shfl_xor(v, 8, 32);
    v += __shfl_xor(v, 4, 32);
    v += __shfl_xor(v, 2, 32);
    v += __shfl_xor(v, 1, 32);
    return v;
}

__global__ __launch_bounds__(256) void nnue_fused_kernel(
    const float* __restrict__ values,        // [B, 32]
    const int*   __restrict__ stm_indices,   // [B, 32]
    const int*   __restrict__ nstm_indices,  // [B, 32]
    const float* __restrict__ ft_w,          // [49152, 512]
    const float* __restrict__ ft_b,          // [512]
    const float* __restrict__ fft_w,         // [768, 512]
    const float* __restrict__ fft_b,         // [512]
    const float* __restrict__ out_w,         // [1, 1024]
    const float* __restrict__ out_b,         // [1]
    float*       __restrict__ out)           // [B, 1]
{
    __shared__ __align__(16) float lds_outw[2 * FT_OUT]; // 1024 floats = 4 KB
    __shared__ __align__(16) float lds_bias[FT_OUT];     // ft_b + fft_b

    const int tid = threadIdx.x;

    // ---- Stage out_w (4 KB) into LDS via CDNA5 async global->LDS copy ----
    // 256 threads x one b128 each covers 1024 floats.
#if __has_builtin(__builtin_amdgcn_global_load_async_to_lds_b128)
    {
        gv4i_t* gsrc = (gv4i_t*)(out_w + tid * 4);
        lv4i_t* ldst = (lv4i_t*)(&lds_outw[tid * 4]);
        __builtin_amdgcn_global_load_async_to_lds_b128(gsrc, ldst, 0, 0);
    }
#else
    *(float4*)&lds_outw[tid * 4] = *(const float4*)(out_w + tid * 4);
#endif

    // ---- Stage combined bias (ft_b + fft_b) into LDS ----
    if (tid < FT_OUT / 4) {
        float4 a = *(const float4*)(ft_b + tid * 4);
        float4 c = *(const float4*)(fft_b + tid * 4);
        float4 s;
        s.x = a.x + c.x; s.y = a.y + c.y; s.z = a.z + c.z; s.w = a.w + c.w;
        *(float4*)&lds_bias[tid * 4] = s;
    }

#if __has_builtin(__builtin_amdgcn_global_load_async_to_lds_b128)
#  if __has_builtin(__builtin_amdgcn_s_wait_asynccnt)
    __builtin_amdgcn_s_wait_asynccnt(0);
#  else
    asm volatile("s_wait_asynccnt 0" ::: "memory");
#  endif
#endif
    __syncthreads();

    // ---- One wave32 per batch element ----
    const int lane = tid & 31;
    const int wave = tid >> 5;
    const int b    = blockIdx.x * WAVES_PER_BLOCK + wave;

    // Lane k holds feature k's value and indices for this batch element.
    const float    vv  = values[b * K_FEAT + lane];
    const int      i1v = stm_indices[b * K_FEAT + lane];
    const int      i2v = nstm_indices[b * K_FEAT + lane];
    const unsigned vb  = __float_as_uint(vv);

    // acc1 -> hidden[0:512] (stm), acc2 -> hidden[512:1024] (nstm).
    // Lane covers dims d = j*128 + lane*4 + {0..3}, j = 0..3.
    float4 acc1[4], acc2[4];
#pragma unroll
    for (int j = 0; j < 4; ++j) {
        float4 bias = *(const float4*)&lds_bias[j * 128 + lane * 4];
        acc1[j] = bias;
        acc2[j] = bias;
    }

#pragma unroll 4
    for (int k = 0; k < K_FEAT; ++k) {
        // Prefetch ft_w rows 4 features ahead: 32 lanes x 64B covers the 2KB row.
        {
            const int kp = (k + 4) & (K_FEAT - 1);
            const int p1 = __builtin_amdgcn_readlane(i1v, kp);
            const int p2 = __builtin_amdgcn_readlane(i2v, kp);
            __builtin_prefetch(ft_w + (size_t)p1 * FT_OUT + lane * 16, 0, 0);
            __builtin_prefetch(ft_w + (size_t)p2 * FT_OUT + lane * 16, 0, 0);
        }

        const float vk = __uint_as_float(__builtin_amdgcn_readlane(vb, k));
        const int   i1 = __builtin_amdgcn_readlane(i1v, k);
        const int   i2 = __builtin_amdgcn_readlane(i2v, k);

        const float* __restrict__ r1 = ft_w  + (size_t)i1 * FT_OUT;           // stm ft
        const float* __restrict__ r2 = ft_w  + (size_t)i2 * FT_OUT;           // nstm ft
        const float* __restrict__ r3 = fft_w + (size_t)(i1 % N_FFT) * FT_OUT; // stm fft
        const float* __restrict__ r4 = fft_w + (size_t)(i2 % N_FFT) * FT_OUT; // nstm fft

#pragma unroll
        for (int j = 0; j < 4; ++j) {
            const int off = j * 128 + lane * 4;     // 512B-contiguous per instruction
            float4 w1 = *(const float4*)(r1 + off);
            float4 w3 = *(const float4*)(r3 + off);
            float4 w2 = *(const float4*)(r2 + off);
            float4 w4 = *(const float4*)(r4 + off);
            fma4(acc1[j], vk, w1);
            fma4(acc1[j], vk, w3);
            fma4(acc2[j], vk, w2);
            fma4(acc2[j], vk, w4);
        }
    }

    // ---- clamp, output projection, wave reduction, sigmoid ----
    float partial = 0.0f;
#pragma unroll
    for (int j = 0; j < 4; ++j) {
        const int off = j * 128 + lane * 4;
        float4 h1 = clamp01_4(acc1[j]);
        float4 h2 = clamp01_4(acc2[j]);
        float4 w1 = *(const float4*)&lds_outw[off];
        float4 w2 = *(const float4*)&lds_outw[FT_OUT + off];
        partial += dot4(h1, w1);
        partial += dot4(h2, w2);
    }

    partial = wave32_reduce_add(partial);

    if (lane == 0) {
        const float x = partial + out_b[0];
        out[b] = 1.0f / (1.0f + expf(-x));
    }
}

extern "C" void kernel_launch(void* const* d_in, const int* in_sizes, int n_in,
                              void* d_out, int out_size, void* d_ws, size_t ws_size,
                              hipStream_t stream) {
    const float* values = (const float*)d_in[0];
    const int*   stm    = (const int*)d_in[1];
    const int*   nstm   = (const int*)d_in[2];
    const float* ft_w   = (const float*)d_in[3];
    const float* ft_b   = (const float*)d_in[4];
    const float* fft_w  = (const float*)d_in[5];
    const float* fft_b  = (const float*)d_in[6];
    const float* out_w  = (const float*)d_in[7];
    const float* out_b  = (const float*)d_in[8];
    float* out = (float*)d_out;

    const int B = in_sizes[0] / K_FEAT;             // 8192
    dim3 grid(B / WAVES_PER_BLOCK);                 // 1024 blocks
    dim3 block(32 * WAVES_PER_BLOCK);               // 256 threads (8 waves)
    nnue_fused_kernel<<<grid, block, 0, stream>>>(
        values, stm, nstm, ft_w, ft_b, fft_w, fft_b, out_w, out_b, out);
}